// ResidualGAT_15547781612259
// MI455X (gfx1250) — compile-verified
//
#include <hip/hip_runtime.h>
#include <stdint.h>

typedef __attribute__((ext_vector_type(2))) float v2f;
typedef __attribute__((ext_vector_type(8))) float v8f;

#define C_IN      128
#define HEADS     4
#define COLS      (HEADS * C_IN)   // 512
#define NEG_SLOPE 0.2f

// ---------------------------------------------------------------------------
// Ordered-uint encoding of float for atomicMax-based segment max.
// Monotonic: f1 < f2  <=>  enc(f1) < enc(f2) (unsigned).
// ---------------------------------------------------------------------------
__device__ __forceinline__ unsigned enc_ord(float f) {
  unsigned u = __float_as_uint(f);
  return (u & 0x80000000u) ? ~u : (u | 0x80000000u);
}
__device__ __forceinline__ float dec_ord(unsigned u) {
  return __uint_as_float((u & 0x80000000u) ? (u & 0x7FFFFFFFu) : ~u);
}
#define ENC_NEG_INF 0x007FFFFFu   // enc_ord(-inf)

__device__ __forceinline__ float leaky(float v) {
  return v > 0.f ? v : NEG_SLOPE * v;
}

// ---------------------------------------------------------------------------
// GEMM: XH[n, h*128+c] = sum_i X[n,i] * W[h,i,c]
// One wave32 per 16x64 output tile: 4 accumulators share one A fragment per
// k-step -> 4 x V_WMMA_F32_16X16X4_F32 per A load (fp32 pipe, full precision).
// A-frag layout (ISA 7.12.2, 32-bit A 16x4): lanes 0-15 -> K=k0,k0+1;
// lanes 16-31 -> K=k0+2,k0+3, row M = lane&15.
// C/D layout: VGPR v holds rows (v, v+8) split across lane halves, N = lane&15.
// Requires N % 16 == 0 (holds: N = 20000); COLS = 512 -> 8 column tiles of 64,
// which never straddle a head boundary (64 | 128).
// ---------------------------------------------------------------------------
__global__ void gemm_xh(const float* __restrict__ X, const float* __restrict__ W,
                        float* __restrict__ XH, int N) {
  int wave = (blockIdx.x * blockDim.x + threadIdx.x) >> 5;
  int lane = threadIdx.x & 31;
  const int tilesN = COLS / 64;          // 8
  int tm = wave / tilesN;
  int tn = wave - tm * tilesN;
  if (tm * 16 >= N) return;

  const int m_base = tm * 16;
  const int n_base = tn * 64;
  const int head   = n_base >> 7;
  const int cb     = n_base & 127;
  const float* __restrict__ Wh = W + (size_t)head * C_IN * C_IN; // [128x128] (i,c)

  const int halfsel = lane >> 4;         // 0: lanes 0-15, 1: lanes 16-31
  const int l15     = lane & 15;
  const int arow    = m_base + l15;

  v8f acc0 = {}, acc1 = {}, acc2 = {}, acc3 = {};
  #pragma unroll 2
  for (int k0 = 0; k0 < C_IN; k0 += 4) {
    const int ka = k0 + halfsel * 2;
    v2f a;
    a.x = X[(size_t)arow * C_IN + ka];
    a.y = X[(size_t)arow * C_IN + ka + 1];
    const float* __restrict__ w0 = Wh + (size_t)ka * C_IN + cb + l15;
    const float* __restrict__ w1 = w0 + C_IN;           // row ka+1
    v2f b0, b1, b2, b3;
    b0.x = w0[0];  b0.y = w1[0];
    b1.x = w0[16]; b1.y = w1[16];
    b2.x = w0[32]; b2.y = w1[32];
    b3.x = w0[48]; b3.y = w1[48];
    acc0 = __builtin_amdgcn_wmma_f32_16x16x4_f32(false, a, false, b0, (short)0, acc0, false, false);
    acc1 = __builtin_amdgcn_wmma_f32_16x16x4_f32(false, a, false, b1, (short)0, acc1, false, false);
    acc2 = __builtin_amdgcn_wmma_f32_16x16x4_f32(false, a, false, b2, (short)0, acc2, false, false);
    acc3 = __builtin_amdgcn_wmma_f32_16x16x4_f32(false, a, false, b3, (short)0, acc3, false, false);
  }

  float* __restrict__ out0 =
      XH + (size_t)(m_base + halfsel * 8) * COLS + n_base + l15;
  #pragma unroll
  for (int v = 0; v < 8; ++v) {
    float* o = out0 + (size_t)v * COLS;
    o[0]  = acc0[v];
    o[16] = acc1[v];
    o[32] = acc2[v];
    o[48] = acc3[v];
  }
}

// ---------------------------------------------------------------------------
// Attention projections: als[n,h] = <xh[n,h,:], a_src[h,:]>, same for ald.
// One wave per (n, h); 4 elems/lane then shfl_xor tree reduce (wave32).
// ---------------------------------------------------------------------------
__global__ void att_dots(const float* __restrict__ XH,
                         const float* __restrict__ ASRC,
                         const float* __restrict__ ADST,
                         float* __restrict__ ALS, float* __restrict__ ALD, int N) {
  int w    = (blockIdx.x * blockDim.x + threadIdx.x) >> 5;
  int lane = threadIdx.x & 31;
  int n = w >> 2, h = w & 3;
  if (n >= N) return;
  const float* __restrict__ row = XH + (size_t)n * COLS + h * C_IN;
  const float* __restrict__ as  = ASRC + h * C_IN;
  const float* __restrict__ ad  = ADST + h * C_IN;
  float s0 = 0.f, s1 = 0.f;
  #pragma unroll
  for (int j = 0; j < C_IN; j += 32) {
    float v = row[j + lane];
    s0 += v * as[j + lane];
    s1 += v * ad[j + lane];
  }
  #pragma unroll
  for (int off = 16; off; off >>= 1) {
    s0 += __shfl_xor(s0, off, 32);
    s1 += __shfl_xor(s1, off, 32);
  }
  if (lane == 0) { ALS[n * 4 + h] = s0; ALD[n * 4 + h] = s1; }
}

// ---------------------------------------------------------------------------
// Init per-dst max (enc(-inf)) and denominators.
// ---------------------------------------------------------------------------
__global__ void init_md(unsigned* __restrict__ MU, float* __restrict__ DEN, int n) {
  int i = blockIdx.x * blockDim.x + threadIdx.x;
  if (i < n) { MU[i] = ENC_NEG_INF; DEN[i] = 0.f; }
}

__global__ void zero_f4(float4* __restrict__ p, long long n4) {
  long long i = (long long)blockIdx.x * blockDim.x + threadIdx.x;
  if (i < n4) p[i] = make_float4(0.f, 0.f, 0.f, 0.f);
}

__device__ __forceinline__ void edge_sd(const int* __restrict__ EI, int E, int e,
                                        int& s, int& d) {
  if (e < E) { s = EI[e]; d = EI[E + e]; }
  else       { s = d = e - E; }           // self-loops appended after real edges
}

// ---------------------------------------------------------------------------
// Pass 1: segment max over incoming edges (ordered-uint atomicMax).
// ---------------------------------------------------------------------------
__global__ void edge_max(const int* __restrict__ EI, int E, int Etot,
                         const float* __restrict__ ALS, const float* __restrict__ ALD,
                         unsigned* __restrict__ MU) {
  int e = blockIdx.x * blockDim.x + threadIdx.x;
  if (e >= Etot) return;
  int s, d; edge_sd(EI, E, e, s, d);
  #pragma unroll
  for (int h = 0; h < HEADS; ++h) {
    float v = leaky(ALS[s * 4 + h] + ALD[d * 4 + h]);
    atomicMax(&MU[d * 4 + h], enc_ord(v));
  }
}

// ---------------------------------------------------------------------------
// Pass 2: ex = exp(e - max); store per-edge, atomicAdd into denominator.
// ---------------------------------------------------------------------------
__global__ void edge_exp(const int* __restrict__ EI, int E, int Etot,
                         const float* __restrict__ ALS, const float* __restrict__ ALD,
                         const unsigned* __restrict__ MU,
                         float* __restrict__ EX, float* __restrict__ DEN) {
  int e = blockIdx.x * blockDim.x + threadIdx.x;
  if (e >= Etot) return;
  int s, d; edge_sd(EI, E, e, s, d);
  #pragma unroll
  for (int h = 0; h < HEADS; ++h) {
    float v  = leaky(ALS[s * 4 + h] + ALD[d * 4 + h]);
    float ex = __expf(v - dec_ord(MU[d * 4 + h]));
    EX[(size_t)e * 4 + h] = ex;
    atomicAdd(&DEN[d * 4 + h], ex);
  }
}

// ---------------------------------------------------------------------------
// Pass 3: scatter alpha * xh[src] into AGG[dst]. One wave per edge handles
// all 4 heads: ex[e,0:4] and den[dst,0:4] come in as single b128 loads,
// each head's 128 channels move as float4/lane (xh L2-resident).
// ---------------------------------------------------------------------------
__global__ void scatter_msg(const int* __restrict__ EI, int E, int Etot,
                            const float* __restrict__ EX, const float* __restrict__ DEN,
                            const float* __restrict__ XH, float* __restrict__ AGG) {
  int e    = (blockIdx.x * blockDim.x + threadIdx.x) >> 5;
  int lane = threadIdx.x & 31;
  if (e >= Etot) return;
  int s, d; edge_sd(EI, E, e, s, d);

  const float4 exv  = *reinterpret_cast<const float4*>(EX  + (size_t)e * 4);
  const float4 denv = *reinterpret_cast<const float4*>(DEN + (size_t)d * 4);
  const float alpha[HEADS] = { exv.x / denv.x, exv.y / denv.y,
                               exv.z / denv.z, exv.w / denv.w };

  const float* __restrict__ srcRow = XH  + (size_t)s * COLS + lane * 4;
  float* __restrict__       dstRow = AGG + (size_t)d * COLS + lane * 4;
  #pragma unroll
  for (int h = 0; h < HEADS; ++h) {
    const float4 v = *reinterpret_cast<const float4*>(srcRow + h * C_IN);
    float* dst = dstRow + h * C_IN;
    const float a = alpha[h];
    atomicAdd(dst + 0, a * v.x);
    atomicAdd(dst + 1, a * v.y);
    atomicAdd(dst + 2, a * v.z);
    atomicAdd(dst + 3, a * v.w);
  }
}

// ---------------------------------------------------------------------------
// Finalize: head mean + bias; optionally accumulate onto existing OUT
// (used to fuse the residual add of the last layer).
// ---------------------------------------------------------------------------
__global__ void finalize(const float* __restrict__ AGG, const float* __restrict__ B,
                         float* __restrict__ OUT, int N, int addPrev) {
  int i = blockIdx.x * blockDim.x + threadIdx.x;
  if (i >= N * C_IN) return;
  int n = i >> 7, c = i & 127;
  const float* g = AGG + (size_t)n * COLS + c;
  float v = 0.25f * (g[0] + g[128] + g[256] + g[384]) + B[c];
  if (addPrev) v += OUT[i];
  OUT[i] = v;
}

// ---------------------------------------------------------------------------
// Host-side: one GATConv layer.
// ---------------------------------------------------------------------------
static void run_gat_layer(const float* x_in, const float* W, const float* as,
                          const float* ad, const float* b, const int* EI,
                          int N, int E, int Etot, float* OUT, int addPrev,
                          float* xh, float* agg, float* als, float* ald,
                          unsigned* mu, float* den, float* ex, hipStream_t stream) {
  const int tiles = (N / 16) * (COLS / 64);                 // waves for GEMM
  gemm_xh   <<<(tiles + 7) / 8, 256, 0, stream>>>(x_in, W, xh, N);
  att_dots  <<<(N * 4 + 7) / 8, 256, 0, stream>>>(xh, as, ad, als, ald, N);
  init_md   <<<(N * 4 + 255) / 256, 256, 0, stream>>>(mu, den, N * 4);
  {
    long long n4 = (long long)N * COLS / 4;
    zero_f4 <<<(unsigned)((n4 + 255) / 256), 256, 0, stream>>>((float4*)agg, n4);
  }
  edge_max  <<<(Etot + 255) / 256, 256, 0, stream>>>(EI, E, Etot, als, ald, mu);
  edge_exp  <<<(Etot + 255) / 256, 256, 0, stream>>>(EI, E, Etot, als, ald, mu, ex, den);
  scatter_msg<<<(Etot + 7) / 8, 256, 0, stream>>>(EI, E, Etot, ex, den, xh, agg);
  finalize  <<<(N * C_IN + 255) / 256, 256, 0, stream>>>(agg, b, OUT, N, addPrev);
}

// ---------------------------------------------------------------------------
// Entry point. Input order (setup_inputs): x, edge_index,
// then (W, a_src, a_dst, b) for layers '1', '2', 'r'.
// ---------------------------------------------------------------------------
extern "C" void kernel_launch(void* const* d_in, const int* in_sizes, int n_in,
                              void* d_out, int out_size, void* d_ws, size_t ws_size,
                              hipStream_t stream) {
  const float* x   = (const float*)d_in[0];
  const int*   EI  = (const int*)d_in[1];
  const float* W1  = (const float*)d_in[2];
  const float* as1 = (const float*)d_in[3];
  const float* ad1 = (const float*)d_in[4];
  const float* b1  = (const float*)d_in[5];
  const float* W2  = (const float*)d_in[6];
  const float* as2 = (const float*)d_in[7];
  const float* ad2 = (const float*)d_in[8];
  const float* b2  = (const float*)d_in[9];
  const float* Wr  = (const float*)d_in[10];
  const float* asr = (const float*)d_in[11];
  const float* adr = (const float*)d_in[12];
  const float* br  = (const float*)d_in[13];

  const int N    = in_sizes[0] / C_IN;   // 20000
  const int E    = in_sizes[1] / 2;      // 320000
  const int Etot = E + N;                // + self-loops

  // Workspace carve-up (floats). Total ~99 MB.
  float* p   = (float*)d_ws;
  float* xh  = p; p += (size_t)N * COLS;       // [N, H*C]
  float* agg = p; p += (size_t)N * COLS;       // [N, H*C] accumulator
  float* als = p; p += (size_t)N * HEADS;
  float* ald = p; p += (size_t)N * HEADS;
  unsigned* mu = (unsigned*)p; p += (size_t)N * HEADS;
  float* den = p; p += (size_t)N * HEADS;
  float* ex  = p; p += (size_t)Etot * HEADS;
  float* h1  = p; p += (size_t)N * C_IN;       // layer-1 output
  float* out = (float*)d_out;

  // res = GAT_r(x)            -> d_out   (overwrite)
  run_gat_layer(x,  Wr, asr, adr, br, EI, N, E, Etot, out, 0,
                xh, agg, als, ald, mu, den, ex, stream);
  // h1 = GAT_1(x)             -> ws
  run_gat_layer(x,  W1, as1, ad1, b1, EI, N, E, Etot, h1, 0,
                xh, agg, als, ald, mu, den, ex, stream);
  // out = GAT_2(h1) + res     -> d_out   (fused residual add)
  run_gat_layer(h1, W2, as2, ad2, b2, EI, N, E, Etot, out, 1,
                xh, agg, als, ald, mu, den, ex, stream);
}